// COOTensorProduct_11536282157111
// MI455X (gfx1250) — compile-verified
//
#include <hip/hip_runtime.h>

typedef __attribute__((ext_vector_type(2))) float v2f;
typedef __attribute__((ext_vector_type(4))) float v4f;
typedef __attribute__((ext_vector_type(8))) float v8f;
typedef int vi4 __attribute__((vector_size(16)));   // matches builtin param pointee

#define B_ROWS   4096   // batch
#define DIM      64     // dim1 == dim2
#define K_TOT    4096   // dim1*dim2
#define H_OUT    4096   // cb_height
#define BM       128    // batch rows per block
#define BN       128    // cb rows per block
#define KC       64     // K chunk (aligned so i = k/64 is constant per chunk)
#define LDP      68     // padded LDS row stride (floats): stride%64==4 -> conflict-free b64 frag loads
#define NCHUNK   (K_TOT / KC)

#if __has_builtin(__builtin_amdgcn_global_load_async_to_lds_b128) && \
    __has_builtin(__builtin_amdgcn_s_wait_asynccnt)
#define USE_ASYNC 1
#else
#define USE_ASYNC 0
#endif

#if USE_ASYNC
__device__ __forceinline__ void async_cp16(const float* g, float* l) {
    // GLOBAL_LOAD_ASYNC_TO_LDS_B128: per-lane 16B global->LDS, tracked by ASYNCcnt
    __builtin_amdgcn_global_load_async_to_lds_b128(
        (__attribute__((address_space(1))) vi4*)g,
        (__attribute__((address_space(3))) vi4*)l,
        0, 0);
}
#endif

__global__ __launch_bounds__(256)
void coo_tp_wmma_f32(const float* __restrict__ in1,
                     const float* __restrict__ in2,
                     const float* __restrict__ cb,
                     float* __restrict__ out)
{
    extern __shared__ float smem[];
    float* s_in1 = smem;                  // [BM][LDP]
    float* s_in2 = smem + BM * LDP;       // [BM][LDP]
    float* s_cb  = smem + 2 * BM * LDP;   // [2][BN][LDP]

    const int tid  = threadIdx.x;
    const int lane = tid & 31;
    const int wv   = tid >> 5;   // wave 0..7
    const int wm   = wv >> 2;    // 0..1 : 64-row band
    const int wn   = wv & 3;     // 0..3 : 32-col band
    const int mBase = blockIdx.y * BM;
    const int nBase = blockIdx.x * BN;

#if USE_ASYNC
    // ---- kick off async DMA of cb chunk 0 into LDS buffer 0 ----
    #pragma unroll
    for (int s = 0; s < 8; ++s) {
        int q = tid + 256 * s;            // 0..2047 float4 slots (128 rows x 16 float4)
        int row = q >> 4, c4 = q & 15;
        async_cp16(cb + (size_t)(nBase + row) * K_TOT + c4 * 4,
                   s_cb + row * LDP + c4 * 4);
    }
#endif

    // ---- stage in1/in2 row blocks (once), overlapped with chunk-0 DMA ----
    #pragma unroll
    for (int s = 0; s < 8; ++s) {
        int q   = tid + 256 * s;
        int row = q >> 4;
        int c4  = q & 15;
        v4f a = *(const v4f*)(in1 + (size_t)(mBase + row) * DIM + c4 * 4);
        v4f b = *(const v4f*)(in2 + (size_t)(mBase + row) * DIM + c4 * 4);
        *(v4f*)(s_in1 + row * LDP + c4 * 4) = a;
        *(v4f*)(s_in2 + row * LDP + c4 * 4) = b;
    }

#if !USE_ASYNC
    // ---- fallback: register-stage cb chunk 0 ----
    v4f pf[8];
    #pragma unroll
    for (int s = 0; s < 8; ++s) {
        int q = tid + 256 * s;
        int row = q >> 4, c4 = q & 15;
        pf[s] = *(const v4f*)(cb + (size_t)(nBase + row) * K_TOT + c4 * 4);
    }
    #pragma unroll
    for (int s = 0; s < 8; ++s) {
        int q = tid + 256 * s;
        int row = q >> 4, c4 = q & 15;
        *(v4f*)(s_cb + row * LDP + c4 * 4) = pf[s];
    }
#else
    __builtin_amdgcn_s_wait_asynccnt(0);
#endif
    __syncthreads();

    v8f zero = {0.f, 0.f, 0.f, 0.f, 0.f, 0.f, 0.f, 0.f};
    v8f acc[4][2];
    #pragma unroll
    for (int mt = 0; mt < 4; ++mt)
        #pragma unroll
        for (int nt = 0; nt < 2; ++nt)
            acc[mt][nt] = zero;

    const int r15   = lane & 15;        // M (or N) index within a 16-tile
    const int khalf = (lane >> 4) << 1; // K pair base: 0 or 2 (ISA 16x4 f32 A/B layout)

    for (int ic = 0; ic < NCHUNK; ++ic) {
        float* cur = s_cb + (ic & 1) * (BN * LDP);

        // stage next cb chunk while computing on current
        if (ic + 1 < NCHUNK) {
            float* nxt = s_cb + ((ic + 1) & 1) * (BN * LDP);
            (void)nxt;
#if USE_ASYNC
            #pragma unroll
            for (int s = 0; s < 8; ++s) {
                int q = tid + 256 * s;
                int row = q >> 4, c4 = q & 15;
                async_cp16(cb + (size_t)(nBase + row) * K_TOT + (ic + 1) * KC + c4 * 4,
                           nxt + row * LDP + c4 * 4);
            }
#else
            #pragma unroll
            for (int s = 0; s < 8; ++s) {
                int q = tid + 256 * s;
                int row = q >> 4, c4 = q & 15;
                pf[s] = *(const v4f*)(cb + (size_t)(nBase + row) * K_TOT
                                         + (ic + 1) * KC + c4 * 4);
            }
#endif
        }

        // per-chunk in1 scalars: outer[b, ic*64 + t] = in1[b, ic] * in2[b, t]
        float a1[4];
        #pragma unroll
        for (int mt = 0; mt < 4; ++mt)
            a1[mt] = s_in1[(wm * 64 + mt * 16 + r15) * LDP + ic];

        #pragma unroll
        for (int t4 = 0; t4 < KC; t4 += 4) {
            // B fragments: B[k,n] = cb[nBase+n, ic*64 + t4 + k]
            v2f bf0 = *(const v2f*)(cur + (wn * 32 +  0 + r15) * LDP + t4 + khalf);
            v2f bf1 = *(const v2f*)(cur + (wn * 32 + 16 + r15) * LDP + t4 + khalf);
            #pragma unroll
            for (int mt = 0; mt < 4; ++mt) {
                // synthesize A fragment from rank-1 structure (no outer matrix in memory)
                v2f v = *(const v2f*)(s_in2 + (wm * 64 + mt * 16 + r15) * LDP + t4 + khalf);
                v2f af;
                af.x = a1[mt] * v.x;
                af.y = a1[mt] * v.y;
                acc[mt][0] = __builtin_amdgcn_wmma_f32_16x16x4_f32(
                    false, af, false, bf0, (short)0, acc[mt][0], false, false);
                acc[mt][1] = __builtin_amdgcn_wmma_f32_16x16x4_f32(
                    false, af, false, bf1, (short)0, acc[mt][1], false, false);
            }
        }

#if USE_ASYNC
        if (ic + 1 < NCHUNK)
            __builtin_amdgcn_s_wait_asynccnt(0);   // own DMAs into nxt complete
        __syncthreads();                            // everyone's DMAs complete
#else
        __syncthreads();
        if (ic + 1 < NCHUNK) {
            float* nxt = s_cb + ((ic + 1) & 1) * (BN * LDP);
            #pragma unroll
            for (int s = 0; s < 8; ++s) {
                int q = tid + 256 * s;
                int row = q >> 4, c4 = q & 15;
                *(v4f*)(nxt + row * LDP + c4 * 4) = pf[s];
            }
        }
        __syncthreads();
#endif
    }

    // ---- store: C/D layout: vgpr v, lanes 0-15 -> M=v, lanes 16-31 -> M=v+8 ----
    #pragma unroll
    for (int mt = 0; mt < 4; ++mt) {
        #pragma unroll
        for (int nt = 0; nt < 2; ++nt) {
            int col = nBase + wn * 32 + nt * 16 + r15;
            int rb  = mBase + wm * 64 + mt * 16 + (lane >> 4) * 8;
            #pragma unroll
            for (int v = 0; v < 8; ++v)
                out[(size_t)(rb + v) * H_OUT + col] = acc[mt][nt][v];
        }
    }
}

extern "C" void kernel_launch(void* const* d_in, const int* in_sizes, int n_in,
                              void* d_out, int out_size, void* d_ws, size_t ws_size,
                              hipStream_t stream)
{
    const float* in1 = (const float*)d_in[0];
    const float* in2 = (const float*)d_in[1];
    const float* cb  = (const float*)d_in[2];
    float* out = (float*)d_out;

    dim3 grid(H_OUT / BN, B_ROWS / BM);   // (32, 32)
    dim3 block(256);                      // 8 waves (wave32)
    size_t shmem = (size_t)(2 * BM * LDP + 2 * BN * LDP) * sizeof(float); // 139264 B
    hipLaunchKernelGGL(coo_tp_wmma_f32, grid, block, shmem, stream, in1, in2, cb, out);
}